// ReNetLayer_63136019251817
// MI455X (gfx1250) — compile-verified
//
#include <hip/hip_runtime.h>
#include <hip/hip_bf16.h>

// ---------------------------------------------------------------------------
// ReNet layer: patch extract + fused-GEMM SRU scan (dir-bidir) -> WMMA fp32
// GEMM (100352x256 @ 256x768) -> second SRU scan -> permuted output.
// ---------------------------------------------------------------------------

typedef float v2f __attribute__((ext_vector_type(2)));
typedef float v8f __attribute__((ext_vector_type(8)));

#define HP 112
#define WP 112
#define NB 8          // batch
#define DHID 128      // D per direction
#define E2 256        // 2*D
#define M2 100352     // Hp*Wp*B rows of GEMM2
#define K2 256
#define N2C 768       // 2*D*3 columns of GEMM2
#define SCALE_X 1.41421356237309504880f

__device__ __forceinline__ float sigmoidf_fast(float v) {
    return 1.0f / (1.0f + __expf(-v));
}

// ---------------------------------------------------------------------------
// Kernel 1: SRU-1 with U1 = patch @ W1 computed on the fly (K=12, k=4).
// One block per n1 = hp*B + b; thread = e = dir*128 + d.
// Writes h1 directly in a2 layout: h1[hp*896 + w*8 + b][e].
// ---------------------------------------------------------------------------
__global__ __launch_bounds__(256) void renet_scan1(
    const float* __restrict__ x,   // (B,3,224,224)
    const float* __restrict__ W1,  // (12, 1024)
    const float* __restrict__ wc1, // (512)
    const float* __restrict__ b1,  // (512)
    float* __restrict__ h1)        // (Hp*Wp*B, 256) in a2 layout
{
    const int n1  = blockIdx.x;     // 0..895 = hp*8 + b
    const int hp  = n1 >> 3;
    const int b   = n1 & 7;
    const int tid = threadIdx.x;    // e
    const int dir = tid >> 7;
    const int d   = tid & 127;

    // Preload the 12x4 W1 coefficients this (dir,d) needs into registers.
    float wco[12][4];
    const int colbase = dir * 512 + d * 4;
#pragma unroll
    for (int f = 0; f < 12; ++f) {
#pragma unroll
        for (int kk = 0; kk < 4; ++kk)
            wco[f][kk] = W1[f * 1024 + colbase + kk];
    }
    const float vf = wc1[tid];
    const float vr = wc1[256 + tid];
    const float bf = b1[tid];
    const float br = b1[256 + tid];

    const size_t xbase = (size_t)b * 3 * 224 * 224;
    float c = 0.0f;

    for (int t = 0; t < WP; ++t) {
        const int w = dir ? (WP - 1 - t) : t;
        float u0 = 0.f, u1 = 0.f, u2 = 0.f, u3 = 0.f;
#pragma unroll
        for (int ch = 0; ch < 3; ++ch) {
#pragma unroll
            for (int wh = 0; wh < 2; ++wh) {
#pragma unroll
                for (int ww = 0; ww < 2; ++ww) {
                    const float xv =
                        x[xbase + ((size_t)ch * 224 + (2 * hp + wh)) * 224 + (2 * w + ww)];
                    const int f = ch * 4 + wh * 2 + ww;
                    u0 = fmaf(xv, wco[f][0], u0);
                    u1 = fmaf(xv, wco[f][1], u1);
                    u2 = fmaf(xv, wco[f][2], u2);
                    u3 = fmaf(xv, wco[f][3], u3);
                }
            }
        }
        const float fg = sigmoidf_fast(u1 + vf * c + bf);
        c = fg * c + (1.0f - fg) * u0;
        const float r = sigmoidf_fast(u2 + vr * c + br);
        const float h = r * c + (1.0f - r) * u3;
        h1[((size_t)hp * (WP * NB) + (size_t)w * NB + b) * E2 + tid] = h;
    }
}

// ---------------------------------------------------------------------------
// Kernel 2: U2 = h1(a2 layout, 100352x256) @ W2 (256x768) via f32 WMMA.
// Block: 256 threads = 8 waves; block tile 64x64; wave tile 16x32.
// V_WMMA_F32_16X16X4_F32: A 16x4 (2 VGPR), B 4x16 (2 VGPR), C/D 16x16 (8 VGPR)
// ---------------------------------------------------------------------------
#define LDA 20   // padded LDS stride for A (bank-conflict-free 16-lane reads)
#define LDB 80   // padded LDS stride for B

__global__ __launch_bounds__(256) void renet_gemm2_wmma(
    const float* __restrict__ A,   // (M2, 256)
    const float* __restrict__ B,   // (256, 768)
    float* __restrict__ C)         // (M2, 768)
{
    __shared__ float lA[64 * LDA];
    __shared__ float lB[16 * LDB];

    const int tid  = threadIdx.x;
    const int wave = tid >> 5;
    const int lane = tid & 31;
    const int rowBlk = blockIdx.y * 64;
    const int colBlk = blockIdx.x * 64;
    const int wm = wave >> 1;      // 0..3 -> row sub-tile
    const int wn = wave & 1;       // 0..1 -> col sub-tile (covers 32 cols)

    const int m16   = lane & 15;
    const int khalf = (lane >> 4) * 2;   // lanes 0-15 -> K{0,1}; 16-31 -> K{2,3}

    // Global->LDS staging offsets (float4 per thread per matrix).
    const int la_r = tid >> 2;           // 0..63
    const int la_c = (tid & 3) * 4;      // 0,4,8,12
    const int lb_r = tid >> 4;           // 0..15
    const int lb_c = (tid & 15) * 4;     // 0..60

    v8f acc0 = {0.f, 0.f, 0.f, 0.f, 0.f, 0.f, 0.f, 0.f};
    v8f acc1 = {0.f, 0.f, 0.f, 0.f, 0.f, 0.f, 0.f, 0.f};

    for (int k0 = 0; k0 < K2; k0 += 16) {
        const float4 av = *(const float4*)(A + (size_t)(rowBlk + la_r) * K2 + k0 + la_c);
        *(float4*)(&lA[la_r * LDA + la_c]) = av;
        const float4 bv = *(const float4*)(B + (size_t)(k0 + lb_r) * N2C + colBlk + lb_c);
        *(float4*)(&lB[lb_r * LDB + lb_c]) = bv;
        __syncthreads();

#pragma unroll
        for (int kk = 0; kk < 16; kk += 4) {
            v2f a, b0, b1;
            const int ar = (wm * 16 + m16) * LDA + kk + khalf;
            a.x = lA[ar];
            a.y = lA[ar + 1];
            const int br0 = (kk + khalf) * LDB + wn * 32 + m16;
            b0.x = lB[br0];
            b0.y = lB[br0 + LDB];
            b1.x = lB[br0 + 16];
            b1.y = lB[br0 + LDB + 16];
            acc0 = __builtin_amdgcn_wmma_f32_16x16x4_f32(
                false, a, false, b0, (short)0, acc0, false, false);
            acc1 = __builtin_amdgcn_wmma_f32_16x16x4_f32(
                false, a, false, b1, (short)0, acc1, false, false);
        }
        __syncthreads();
    }

    // C/D layout: VGPR v, lanes 0-15: M=v, N=lane; lanes 16-31: M=v+8, N=lane-16
    const int crow = rowBlk + wm * 16 + (lane >> 4) * 8;
    const int ccol = colBlk + wn * 32 + m16;
#pragma unroll
    for (int v = 0; v < 8; ++v) {
        C[(size_t)(crow + v) * N2C + ccol]      = acc0[v];
        C[(size_t)(crow + v) * N2C + ccol + 16] = acc1[v];
    }
}

// ---------------------------------------------------------------------------
// Kernel 3: SRU-2 (k=3, xprime = a2 * sqrt(2)), writes permuted output
// out[b, e, hp, w]. One block per n2 = w*B + b; thread = e.
// ---------------------------------------------------------------------------
__global__ __launch_bounds__(256) void renet_scan2(
    const float* __restrict__ U2,  // (Hp*896, 768)
    const float* __restrict__ h1,  // (Hp*896, 256)  (a2 layout)
    const float* __restrict__ wc2, // (512)
    const float* __restrict__ b2,  // (512)
    float* __restrict__ out)       // (B, 256, Hp, Wp)
{
    const int n2  = blockIdx.x;     // w*8 + b
    const int w   = n2 >> 3;
    const int b   = n2 & 7;
    const int tid = threadIdx.x;    // e
    const int dir = tid >> 7;
    const int d   = tid & 127;

    const float vf = wc2[tid];
    const float vr = wc2[256 + tid];
    const float bf = b2[tid];
    const float br = b2[256 + tid];
    const int col = dir * 384 + d * 3;

    float c = 0.0f;
    for (int t = 0; t < HP; ++t) {
        const int hp = dir ? (HP - 1 - t) : t;
        const size_t row = (size_t)hp * (WP * NB) + n2;
        const float u0 = U2[row * N2C + col];
        const float u1 = U2[row * N2C + col + 1];
        const float u2 = U2[row * N2C + col + 2];
        const float xp = h1[row * E2 + tid] * SCALE_X;

        const float fg = sigmoidf_fast(u1 + vf * c + bf);
        c = fg * c + (1.0f - fg) * u0;
        const float r = sigmoidf_fast(u2 + vr * c + br);
        const float h = r * c + (1.0f - r) * xp;

        out[(((size_t)b * E2 + tid) * HP + hp) * WP + w] = h;
    }
}

// ---------------------------------------------------------------------------
extern "C" void kernel_launch(void* const* d_in, const int* in_sizes, int n_in,
                              void* d_out, int out_size, void* d_ws, size_t ws_size,
                              hipStream_t stream) {
    (void)in_sizes; (void)n_in; (void)out_size; (void)ws_size;

    const float* x   = (const float*)d_in[0];
    const float* W1  = (const float*)d_in[1];
    const float* wc1 = (const float*)d_in[2];
    const float* b1  = (const float*)d_in[3];
    const float* W2  = (const float*)d_in[4];
    const float* wc2 = (const float*)d_in[5];
    const float* b2  = (const float*)d_in[6];
    float* out = (float*)d_out;

    float* h1 = (float*)d_ws;                       // M2 * 256 floats (102.8 MB)
    float* U2 = h1 + (size_t)M2 * E2;               // M2 * 768 floats (308.3 MB)

    renet_scan1<<<dim3(HP * NB), dim3(256), 0, stream>>>(x, W1, wc1, b1, h1);
    renet_gemm2_wmma<<<dim3(N2C / 64, M2 / 64), dim3(256), 0, stream>>>(h1, W2, U2);
    renet_scan2<<<dim3(WP * NB), dim3(256), 0, stream>>>(U2, h1, wc2, b2, out);
}